// variant3_5970004542119
// MI455X (gfx1250) — compile-verified
//
#include <hip/hip_runtime.h>
#include <math.h>

typedef __attribute__((ext_vector_type(2))) float v2f;
typedef __attribute__((ext_vector_type(8))) float v8f;

#define F_IN      128
#define HIDDEN    256
#define HEADS     4
#define CPH       64      // channels per head
#define NEG_SLOPE 0.2f

// ---- order-preserving float <-> uint key for atomicMax on floats ----
__device__ __forceinline__ unsigned enc_f32(float f) {
    unsigned b = __float_as_uint(f);
    return (b & 0x80000000u) ? ~b : (b | 0x80000000u);
}
__device__ __forceinline__ float dec_f32(unsigned k) {
    unsigned b = (k & 0x80000000u) ? (k & 0x7FFFFFFFu) : ~k;
    return __uint_as_float(b);
}

// ---------------- zero scratch (grid-stride) ----------------
__global__ void k_zero(float* p, long n) {
    long i = (long)blockIdx.x * blockDim.x + threadIdx.x;
    long st = (long)gridDim.x * blockDim.x;
    for (; i < n; i += st) p[i] = 0.0f;
}

// ---------------- h = x @ W via V_WMMA_F32_16X16X4_F32 ----------------
// One wave computes one 16x16 tile of h. K=128 -> 32 wmma ops per tile.
// A (16x4 f32): lanes 0-15 hold M=lane, K={0,1}; lanes 16-31 hold M=lane-16, K={2,3}.
// B (4x16 f32): lanes 0-15 hold N=lane, K={0,1} in v0/v1; lanes 16-31 K={2,3}.
// C/D (16x16 f32): vgpr j -> M=j (lanes 0-15) / M=j+8 (lanes 16-31), N=lane&15.
__global__ void k_gemm_wmma(const float* __restrict__ x, const float* __restrict__ W,
                            float* __restrict__ h, int N, int row_tiles) {
    const int wave = threadIdx.x >> 5;
    const int lane = threadIdx.x & 31;
    const int tile = blockIdx.x * 8 + wave;
    const int tile_row = tile >> 4;           // 16 column tiles (HIDDEN/16)
    const int tile_col = tile & 15;
    if (tile_row >= row_tiles) return;        // wave-uniform exit

    const int row_base = tile_row * 16;
    const int col_base = tile_col * 16;
    const int l15 = lane & 15;
    const bool hi = (lane >= 16);

    int row = row_base + l15;
    if (row >= N) row = N - 1;                // clamp loads for tail tile
    const float* xrow = x + (long)row * F_IN + (hi ? 2 : 0);   // 8B aligned
    const float* wcol = W + col_base + l15;

    v8f acc = {};
    #pragma unroll
    for (int k = 0; k < F_IN; k += 4) {
        v2f a = *(const v2f*)(xrow + k);                       // K=k,k+1 (or k+2,k+3)
        const int kb = hi ? (k + 2) : k;
        v2f b;
        b.x = wcol[(long)kb * HIDDEN];
        b.y = wcol[(long)(kb + 1) * HIDDEN];
        acc = __builtin_amdgcn_wmma_f32_16x16x4_f32(
            /*neg_a=*/false, a, /*neg_b=*/false, b,
            /*c_mod=*/(short)0, acc, /*reuse_a=*/false, /*reuse_b=*/false);
    }

    const int m_off = hi ? 8 : 0;
    #pragma unroll
    for (int j = 0; j < 8; ++j) {
        int r = row_base + m_off + j;
        if (r < N) h[(long)r * HIDDEN + col_base + l15] = acc[j];
    }
}

// ---------------- per-node attention logits ----------------
__global__ void k_attn_dots(const float* __restrict__ h,
                            const float* __restrict__ att_src,
                            const float* __restrict__ att_dst,
                            float* __restrict__ a_src, float* __restrict__ a_dst, int N) {
    int t = blockIdx.x * blockDim.x + threadIdx.x;
    if (t >= N * HEADS) return;
    int n = t >> 2, head = t & 3;
    const float* hp = h + (long)n * HIDDEN + head * CPH;
    const float* as = att_src + head * CPH;
    const float* ad = att_dst + head * CPH;
    float s0 = 0.f, s1 = 0.f;
    #pragma unroll 8
    for (int c = 0; c < CPH; ++c) { float v = hp[c]; s0 += v * as[c]; s1 += v * ad[c]; }
    a_src[t] = s0; a_dst[t] = s1;
}

// ---------------- edge pass 1: segment max (via uint keys) ----------------
__global__ void k_edge_max(const int* __restrict__ ei,
                           const float* __restrict__ a_src, const float* __restrict__ a_dst,
                           unsigned* __restrict__ m_key, int E, int E_tot) {
    long t = (long)blockIdx.x * blockDim.x + threadIdx.x;
    if (t >= (long)E_tot * HEADS) return;
    int e = (int)(t >> 2), head = (int)(t & 3);
    int src, dst;
    if (e < E) { src = ei[e]; dst = ei[E + e]; } else { src = dst = e - E; }  // self loop
    float v = a_src[src * HEADS + head] + a_dst[dst * HEADS + head];
    v = (v > 0.f) ? v : NEG_SLOPE * v;
    atomicMax(&m_key[dst * HEADS + head], enc_f32(v));
}

// ---------------- edge pass 2: softmax denominator ----------------
__global__ void k_edge_sum(const int* __restrict__ ei,
                           const float* __restrict__ a_src, const float* __restrict__ a_dst,
                           const unsigned* __restrict__ m_key, float* __restrict__ ssum,
                           int E, int E_tot) {
    long t = (long)blockIdx.x * blockDim.x + threadIdx.x;
    if (t >= (long)E_tot * HEADS) return;
    int e = (int)(t >> 2), head = (int)(t & 3);
    int src, dst;
    if (e < E) { src = ei[e]; dst = ei[E + e]; } else { src = dst = e - E; }
    float v = a_src[src * HEADS + head] + a_dst[dst * HEADS + head];
    v = (v > 0.f) ? v : NEG_SLOPE * v;
    float m = dec_f32(m_key[dst * HEADS + head]);
    atomicAdd(&ssum[dst * HEADS + head], __expf(v - m));
}

// ---------------- edge pass 3: weighted scatter-add (block per edge) ----------------
__global__ void k_edge_aggr(const int* __restrict__ ei,
                            const float* __restrict__ a_src, const float* __restrict__ a_dst,
                            const unsigned* __restrict__ m_key, const float* __restrict__ ssum,
                            const float* __restrict__ h, float* __restrict__ out, int E) {
    int e = blockIdx.x;
    int chan = threadIdx.x;          // 0..255, lane-contiguous channels of one dst row
    int head = chan >> 6;
    int src, dst;
    if (e < E) { src = ei[e]; dst = ei[E + e]; } else { src = dst = e - E; }
    float v = a_src[src * HEADS + head] + a_dst[dst * HEADS + head];
    v = (v > 0.f) ? v : NEG_SLOPE * v;
    float m = dec_f32(m_key[dst * HEADS + head]);
    float alpha = __expf(v - m) / ssum[dst * HEADS + head];
    float val = h[(long)src * HIDDEN + chan] * alpha;
    atomicAdd(&out[(long)dst * HIDDEN + chan], val);   // coalesced across lanes
}

// ---------------- bias + ELU + mean-pool scatter (block per node) ----------------
__global__ void k_elu_pool(const float* __restrict__ out_acc, const float* __restrict__ bias,
                           const int* __restrict__ batch,
                           float* __restrict__ pool, float* __restrict__ counts) {
    int n = blockIdx.x, c = threadIdx.x;
    float v = out_acc[(long)n * HIDDEN + c] + bias[c];
    v = (v > 0.f) ? v : (__expf(v) - 1.f);             // ELU(alpha=1)
    int g = batch[n];
    atomicAdd(&pool[g * HIDDEN + c], v);
    if (c == 0) atomicAdd(&counts[g], 1.0f);
}

// ---------------- final FC over pooled means (block per graph) ----------------
__global__ void k_final_fc(const float* __restrict__ pool, const float* __restrict__ counts,
                           const float* __restrict__ fc_w, const float* __restrict__ fc_b,
                           float* __restrict__ o) {
    __shared__ float red[HIDDEN];
    int g = blockIdx.x, c = threadIdx.x;
    float cnt = counts[g]; cnt = (cnt > 1.f) ? cnt : 1.f;
    red[c] = (pool[g * HIDDEN + c] / cnt) * fc_w[c];
    __syncthreads();
    for (int st = HIDDEN / 2; st > 0; st >>= 1) {
        if (c < st) red[c] += red[c + st];
        __syncthreads();
    }
    if (c == 0) o[g] = red[0] + fc_b[0];
}

extern "C" void kernel_launch(void* const* d_in, const int* in_sizes, int n_in,
                              void* d_out, int out_size, void* d_ws, size_t ws_size,
                              hipStream_t stream) {
    const float* x       = (const float*)d_in[0];
    const int*   ei      = (const int*)  d_in[1];   // [2, E] (src row, then dst row)
    // d_in[2] = edge_attr (unused by the reference)
    const int*   batch   = (const int*)  d_in[3];
    const float* W       = (const float*)d_in[4];
    const float* att_src = (const float*)d_in[5];
    const float* att_dst = (const float*)d_in[6];
    const float* bias    = (const float*)d_in[7];
    const float* fc_w    = (const float*)d_in[8];
    const float* fc_b    = (const float*)d_in[9];
    float* out           = (float*)d_out;

    const int N = in_sizes[0] / F_IN;     // 50000
    const int E = in_sizes[1] / 2;        // 800000
    const int G = out_size;               // 256 graphs
    const int E_tot = E + N;              // with self loops

    // ---- workspace layout (floats) ----
    float* ws      = (float*)d_ws;
    float* h       = ws;                          // N*HIDDEN
    float* a_src   = h + (long)N * HIDDEN;        // N*HEADS
    float* a_dst   = a_src + (long)N * HEADS;     // N*HEADS
    // zero-initialized block starts here (contiguous):
    float* out_acc = a_dst + (long)N * HEADS;     // N*HIDDEN
    float* ssum    = out_acc + (long)N * HIDDEN;  // N*HEADS
    float* pool    = ssum + (long)N * HEADS;      // G*HIDDEN
    float* counts  = pool + (long)G * HIDDEN;     // G
    unsigned* mkey = (unsigned*)(counts + G);     // N*HEADS (key 0 < any real key)
    long zero_elems = (long)N * HIDDEN + (long)N * HEADS + (long)G * HIDDEN + G
                    + (long)N * HEADS;
    (void)ws_size; (void)n_in;

    // 0) zero accumulators + max keys
    k_zero<<<4096, 256, 0, stream>>>(out_acc, zero_elems);

    // 1) h = x @ W (fp32 WMMA)
    int row_tiles = (N + 15) / 16;
    int tiles = row_tiles * (HIDDEN / 16);
    k_gemm_wmma<<<(tiles + 7) / 8, 256, 0, stream>>>(x, W, h, N, row_tiles);

    // 2) attention logits
    k_attn_dots<<<(N * HEADS + 255) / 256, 256, 0, stream>>>(h, att_src, att_dst,
                                                             a_src, a_dst, N);

    // 3) segment max over destinations
    long tmax = (long)E_tot * HEADS;
    k_edge_max<<<(int)((tmax + 255) / 256), 256, 0, stream>>>(ei, a_src, a_dst, mkey, E, E_tot);

    // 4) segment exp-sum
    k_edge_sum<<<(int)((tmax + 255) / 256), 256, 0, stream>>>(ei, a_src, a_dst, mkey, ssum,
                                                              E, E_tot);

    // 5) weighted message scatter-add (one block per edge, coalesced atomics)
    k_edge_aggr<<<E_tot, HIDDEN, 0, stream>>>(ei, a_src, a_dst, mkey, ssum, h, out_acc, E);

    // 6) bias + ELU + mean-pool scatter
    k_elu_pool<<<N, HIDDEN, 0, stream>>>(out_acc, bias, batch, pool, counts);

    // 7) final FC
    k_final_fc<<<G, HIDDEN, 0, stream>>>(pool, counts, fc_w, fc_b, out);
}